// AttentionAggregator_75677323756082
// MI455X (gfx1250) — compile-verified
//
#include <hip/hip_runtime.h>
#include <hip/hip_bf16.h>

#define IN_DIM   128
#define OUT_DIM  64
#define SLOPE    0.1f
#define EPS_F    1e-12f

typedef float v2f __attribute__((ext_vector_type(2)));
typedef float v8f __attribute__((ext_vector_type(8)));

// ---------------------------------------------------------------------------
// Kernel 1: new = x @ W^T + b  via V_WMMA_F32_16X16X4_F32.
// One block = 128 threads = 4 waves; block handles a 16-row stripe of x,
// wave w handles output columns [16w, 16w+16). K loop: 128/4 = 32 WMMAs/wave.
// ---------------------------------------------------------------------------
__global__ void gat_gemm_wmma(const float* __restrict__ x,
                              const float* __restrict__ W,
                              const float* __restrict__ bias,
                              float* __restrict__ newF,
                              int Nnodes) {
    const int lane = threadIdx.x & 31;
    const int wave = threadIdx.x >> 5;       // 0..3 -> which 16-col tile
    const int half = lane >> 4;              // 0: lanes 0-15, 1: lanes 16-31
    const int l15  = lane & 15;

    const int row0 = blockIdx.x * 16;        // 16-row stripe
    const int m    = l15;                    // A-matrix row for this lane
    const int n    = wave * 16 + l15;        // B/C column for this lane

    int arow = row0 + m;
    if (arow >= Nnodes) arow = Nnodes - 1;   // clamp (EXEC must stay all-1s)

    const float* __restrict__ xrow = x + (size_t)arow * IN_DIM;
    const float* __restrict__ wrow = W + (size_t)n    * IN_DIM;

    v8f c = {0.f, 0.f, 0.f, 0.f, 0.f, 0.f, 0.f, 0.f};

#pragma unroll 8
    for (int k0 = 0; k0 < IN_DIM; k0 += 4) {
        const int k = k0 + half * 2;
        v2f a, b;
        // A 16x4: lanes 0-15 carry K={k0,k0+1}, lanes 16-31 K={k0+2,k0+3}
        a.x = xrow[k];
        a.y = xrow[k + 1];
        // B 4x16 (= W^T tile): B(k,n) = W[n*IN_DIM + k], mirrored K striping
        b.x = wrow[k];
        b.y = wrow[k + 1];
        c = __builtin_amdgcn_wmma_f32_16x16x4_f32(
                /*neg_a=*/false, a, /*neg_b=*/false, b,
                /*c_mod=*/(short)0, c, /*reuse_a=*/false, /*reuse_b=*/false);
    }

    const float bn = bias[n];
    if (row0 + 16 <= Nnodes) {
        // Common path: whole tile in-bounds -> 8 unconditional coalesced
        // b32 stores (no exec-mask juggling).
        float* __restrict__ base =
            newF + ((size_t)(row0 + half * 8)) * OUT_DIM + n;
#pragma unroll
        for (int r = 0; r < 8; ++r)
            base[(size_t)r * OUT_DIM] = c[r] + bn;
    } else {
#pragma unroll
        for (int r = 0; r < 8; ++r) {
            const int row = row0 + r + half * 8;
            if (row < Nnodes)
                newF[(size_t)row * OUT_DIM + n] = c[r] + bn;
        }
    }
}

// ---------------------------------------------------------------------------
// Kernel 2: per-node attention scalars  s_src[i] = new[i]·a[0:64],
//                                       s_dst[i] = new[i]·a[64:128].
// One wave (32 lanes) per node, 2 dims per lane, wave shuffle reduction.
// ---------------------------------------------------------------------------
__global__ void gat_scores(const float* __restrict__ newF,
                           const float* __restrict__ a,
                           float* __restrict__ s_src,
                           float* __restrict__ s_dst,
                           int Nnodes) {
    const int lane = threadIdx.x & 31;
    const int node = blockIdx.x * (blockDim.x >> 5) + (threadIdx.x >> 5);
    if (node >= Nnodes) return;

    const float* __restrict__ row = newF + (size_t)node * OUT_DIM;
    const float v0 = row[lane];
    const float v1 = row[lane + 32];
    float ps = v0 * a[lane]           + v1 * a[lane + 32];
    float pd = v0 * a[OUT_DIM + lane] + v1 * a[OUT_DIM + lane + 32];

#pragma unroll
    for (int off = 16; off > 0; off >>= 1) {
        ps += __shfl_xor(ps, off, 32);
        pd += __shfl_xor(pd, off, 32);
    }
    if (lane == 0) {
        s_src[node] = ps;
        s_dst[node] = pd;
    }
}

// ---------------------------------------------------------------------------
// Kernel 3: zero the accumulators (out and row_sum) — run every launch so
// graph replays are deterministic.
// ---------------------------------------------------------------------------
__global__ void gat_zero(float* __restrict__ outAcc,
                         float* __restrict__ row_sum,
                         int nOut, int Nnodes) {
    const int i = blockIdx.x * blockDim.x + threadIdx.x;
    if (i < nOut)   outAcc[i]  = 0.f;
    if (i < Nnodes) row_sum[i] = 0.f;
}

// ---------------------------------------------------------------------------
// Kernel 4: edge phase. One wave per edge. Edge id is forced uniform via
// readfirstlane so the edge indices and node scalars become SMEM s_loads and
// the feature gather/atomics use SGPR-base + lane-offset addressing.
// e = exp(leaky_relu(score)); fire-and-forget global_atomic_add_f32 into
// L2-resident accumulators (64 feature adds + 1 row_sum add per edge).
// ---------------------------------------------------------------------------
__global__ void gat_edges(const long long* __restrict__ ei,   // [2, E] int64
                          const float* __restrict__ newF,
                          const float* __restrict__ s_src,
                          const float* __restrict__ s_dst,
                          float* __restrict__ outAcc,
                          float* __restrict__ row_sum,
                          long long Eedges) {
    const int lane = threadIdx.x & 31;
    // Wave id is uniform across the wave; make it an SGPR value.
    const int wv = __builtin_amdgcn_readfirstlane((int)(threadIdx.x >> 5));
    const long long edge = (long long)blockIdx.x * 8 + wv;
    if (edge >= Eedges) return;

    // Uniform (scalar) loads of the edge endpoints and node scalars.
    const long long s = ei[edge];            // src node (segment id)
    const long long d = ei[Eedges + edge];   // dst node (gathered features)

    const float score = s_src[s] + s_dst[d];           // wave-uniform
    const float lr    = score > 0.f ? score : SLOPE * score;
    const float e     = __expf(lr);

    if (lane == 0) unsafeAtomicAdd(&row_sum[s], e);

    const float* __restrict__ nd = newF   + (size_t)d * OUT_DIM;
    float*       __restrict__ po = outAcc + (size_t)s * OUT_DIM;
    unsafeAtomicAdd(&po[lane],      e * nd[lane]);
    unsafeAtomicAdd(&po[lane + 32], e * nd[lane + 32]);
}

// ---------------------------------------------------------------------------
// Kernel 5: normalize: out[i,:] /= (row_sum[i] + eps)
// ---------------------------------------------------------------------------
__global__ void gat_final(float* __restrict__ outAcc,
                          const float* __restrict__ row_sum,
                          int nOut) {
    const int i = blockIdx.x * blockDim.x + threadIdx.x;
    if (i < nOut)
        outAcc[i] = outAcc[i] / (row_sum[i >> 6] + EPS_F);
}

// ---------------------------------------------------------------------------
extern "C" void kernel_launch(void* const* d_in, const int* in_sizes, int n_in,
                              void* d_out, int out_size, void* d_ws, size_t ws_size,
                              hipStream_t stream) {
    const float*     x  = (const float*)d_in[0];       // [N, 128] f32
    const long long* ei = (const long long*)d_in[1];   // [2, E] int64
    const float*     W  = (const float*)d_in[2];       // [64, 128] f32
    const float*     b  = (const float*)d_in[3];       // [64] f32
    const float*     a  = (const float*)d_in[4];       // [128, 1] f32

    const int       N = in_sizes[0] / IN_DIM;          // 50000
    const long long E = (long long)in_sizes[1] / 2;    // 1600000

    float* out = (float*)d_out;                        // [N, 64] f32

    // workspace layout
    float* ws      = (float*)d_ws;
    float* newF    = ws;                               // N*64 floats
    float* s_src   = newF  + (size_t)N * OUT_DIM;      // N floats
    float* s_dst   = s_src + N;                        // N floats
    float* row_sum = s_dst + N;                        // N floats

    const int nOut = N * OUT_DIM;

    // 1) zero accumulators
    gat_zero<<<(nOut + 255) / 256, 256, 0, stream>>>(out, row_sum, nOut, N);

    // 2) new = x @ W^T + b  (WMMA f32 16x16x4)
    gat_gemm_wmma<<<(N + 15) / 16, 128, 0, stream>>>(x, W, b, newF, N);

    // 3) per-node attention scalars (8 waves/block, 1 node/wave)
    gat_scores<<<(N + 7) / 8, 256, 0, stream>>>(newF, a, s_src, s_dst, N);

    // 4) edge scatter phase (8 waves/block, 1 edge/wave)
    gat_edges<<<(int)((E + 7) / 8), 256, 0, stream>>>(ei, newF, s_src, s_dst,
                                                      out, row_sum, E);

    // 5) normalize
    gat_final<<<(nOut + 255) / 256, 256, 0, stream>>>(out, row_sum, nOut);
}